// SAGENet_33509334843589
// MI455X (gfx1250) — compile-verified
//
#include <hip/hip_runtime.h>

typedef __attribute__((ext_vector_type(2))) float v2f;
typedef __attribute__((ext_vector_type(8))) float v8f;

#define N_SRC0 200000
#define N_FTRS 128
#define N_HID  1500
#define N_CLS  64
#define NE0    1600000
#define NE1    500000
#define RES0   50000
#define RES1   10000

// workspace layout (float offsets)
#define OFF_AGG1 0
#define OFF_CNT1 (OFF_AGG1 + (size_t)RES0 * N_FTRS)   //  6,400,000
#define OFF_AGG2 (OFF_CNT1 + (size_t)RES0)            //  6,450,000
#define OFF_CNT2 (OFF_AGG2 + (size_t)RES1 * N_HID)    // 21,450,000
#define OFF_H    (OFF_CNT2 + (size_t)RES1)            // 21,460,000
// total = 96,460,000 floats ~= 386 MB

__device__ __forceinline__ void atomAdd(float* p, float v) {
  unsafeAtomicAdd(p, v);   // lowers to global_atomic_add_f32
}

// ---- layer-1 mean aggregation: one wave per edge, float4 per lane ----------
__global__ void __launch_bounds__(256)
sage_scatter1(const float* __restrict__ x, const int* __restrict__ src,
              const int* __restrict__ tgt, float* __restrict__ agg,
              float* __restrict__ cnt) {
  int wave = (blockIdx.x * blockDim.x + threadIdx.x) >> 5;
  int lane = threadIdx.x & 31;
  if (wave >= NE0) return;
  int s = src[wave];
  int t = tgt[wave];
  const float4* xr = (const float4*)(x + (size_t)s * N_FTRS);
  float4 v = xr[lane];
  float* dst = agg + (size_t)t * N_FTRS + lane * 4;
  atomAdd(dst + 0, v.x);
  atomAdd(dst + 1, v.y);
  atomAdd(dst + 2, v.z);
  atomAdd(dst + 3, v.w);
  if (lane == 0) atomAdd(cnt + t, 1.0f);
}

// ---- layer-2 mean aggregation: one block per edge --------------------------
__global__ void __launch_bounds__(256)
sage_scatter2(const float* __restrict__ h, const int* __restrict__ src,
              const int* __restrict__ tgt, float* __restrict__ agg,
              float* __restrict__ cnt) {
  int e = blockIdx.x;
  int s = src[e];
  int t = tgt[e];
  const float* hr = h + (size_t)s * N_HID;
  float* ag = agg + (size_t)t * N_HID;
  for (int f = threadIdx.x; f < N_HID; f += blockDim.x)
    atomAdd(ag + f, hr[f]);
  if (threadIdx.x == 0) atomAdd(cnt + t, 1.0f);
}

// ---- cnt -> 1/max(cnt,1) ---------------------------------------------------
__global__ void __launch_bounds__(256)
sage_invcnt(float* __restrict__ c, int n) {
  int i = blockIdx.x * blockDim.x + threadIdx.x;
  if (i < n) c[i] = 1.0f / fmaxf(c[i], 1.0f);
}

// ---- layer 1 GEMM: h = relu(x@W1r + mean@W1n + b1), f32 WMMA ---------------
// one wave per 16x16 tile. M = 50000 (3125 tiles), N padded 1500 -> 94 tiles.
__global__ void __launch_bounds__(32)
sage_gemm1(const float* __restrict__ x, const float* __restrict__ agg,
           const float* __restrict__ inv, const float* __restrict__ Wr,
           const float* __restrict__ Wn, const float* __restrict__ b1,
           float* __restrict__ h) {
  int lane = threadIdx.x;
  int m16  = lane & 15;
  int kg   = lane >> 4;                 // 0: K0/K1, 1: K2/K3 (A layout)
  size_t row = (size_t)blockIdx.x * 16 + m16;
  int col  = blockIdx.y * 16 + m16;
  int colc = col < (N_HID - 1) ? col : (N_HID - 1);  // clamp for safe loads
  const float* xr = x   + row * N_FTRS;
  const float* ar = agg + row * N_FTRS;
  float invc = inv[row];
  v8f acc = {0.f, 0.f, 0.f, 0.f, 0.f, 0.f, 0.f, 0.f};

  for (int k = 0; k < N_FTRS; k += 4) {
    int kk = k + 2 * kg;
    v2f a, b;
    a.x = xr[kk];
    a.y = xr[kk + 1];
    b.x = Wr[(size_t)kk * N_HID + colc];
    b.y = Wr[(size_t)(kk + 1) * N_HID + colc];
    acc = __builtin_amdgcn_wmma_f32_16x16x4_f32(false, a, false, b,
                                                (short)0, acc, false, false);
  }
  for (int k = 0; k < N_FTRS; k += 4) {
    int kk = k + 2 * kg;
    v2f a, b;
    a.x = ar[kk] * invc;
    a.y = ar[kk + 1] * invc;
    b.x = Wn[(size_t)kk * N_HID + colc];
    b.y = Wn[(size_t)(kk + 1) * N_HID + colc];
    acc = __builtin_amdgcn_wmma_f32_16x16x4_f32(false, a, false, b,
                                                (short)0, acc, false, false);
  }
  if (col < N_HID) {
    float bias = b1[col];
    size_t rbase = (size_t)blockIdx.x * 16 + kg * 8;  // C/D layout: +8 for hi half
    for (int r = 0; r < 8; ++r) {
      float v = acc[r] + bias;
      h[(rbase + r) * N_HID + col] = v > 0.f ? v : 0.f;
    }
  }
}

// ---- layer 2 GEMM: logits = h@W2r + mean2@W2n + b2, f32 WMMA ---------------
// M = 10000 (625 tiles), N = 64 (4 tiles), K = 1500 (375 steps of 4). Exact.
__global__ void __launch_bounds__(32)
sage_gemm2(const float* __restrict__ h, const float* __restrict__ agg,
           const float* __restrict__ inv, const float* __restrict__ Wr,
           const float* __restrict__ Wn, const float* __restrict__ b2,
           float* __restrict__ out) {
  int lane = threadIdx.x;
  int m16  = lane & 15;
  int kg   = lane >> 4;
  size_t row = (size_t)blockIdx.x * 16 + m16;
  int col  = blockIdx.y * 16 + m16;
  const float* hr = h   + row * N_HID;
  const float* ar = agg + row * N_HID;
  float invc = inv[row];
  v8f acc = {0.f, 0.f, 0.f, 0.f, 0.f, 0.f, 0.f, 0.f};

  for (int k = 0; k < N_HID; k += 4) {
    int kk = k + 2 * kg;
    v2f a, b;
    a.x = hr[kk];
    a.y = hr[kk + 1];
    b.x = Wr[(size_t)kk * N_CLS + col];
    b.y = Wr[(size_t)(kk + 1) * N_CLS + col];
    acc = __builtin_amdgcn_wmma_f32_16x16x4_f32(false, a, false, b,
                                                (short)0, acc, false, false);
  }
  for (int k = 0; k < N_HID; k += 4) {
    int kk = k + 2 * kg;
    v2f a, b;
    a.x = ar[kk] * invc;
    a.y = ar[kk + 1] * invc;
    b.x = Wn[(size_t)kk * N_CLS + col];
    b.y = Wn[(size_t)(kk + 1) * N_CLS + col];
    acc = __builtin_amdgcn_wmma_f32_16x16x4_f32(false, a, false, b,
                                                (short)0, acc, false, false);
  }
  float bias = b2[col];
  size_t rbase = (size_t)blockIdx.x * 16 + kg * 8;
  for (int r = 0; r < 8; ++r)
    out[(rbase + r) * N_CLS + col] = acc[r] + bias;
}

// ---- row-wise log_softmax over 64 classes: one wave per row ----------------
__global__ void __launch_bounds__(32)
sage_logsoftmax(float* __restrict__ out) {
  int lane = threadIdx.x;
  float* p = out + (size_t)blockIdx.x * N_CLS;
  float v0 = p[lane];
  float v1 = p[lane + 32];
  float m = fmaxf(v0, v1);
  for (int off = 16; off; off >>= 1) m = fmaxf(m, __shfl_xor(m, off, 32));
  float s = expf(v0 - m) + expf(v1 - m);
  for (int off = 16; off; off >>= 1) s += __shfl_xor(s, off, 32);
  float l = m + logf(s);
  p[lane]      = v0 - l;
  p[lane + 32] = v1 - l;
}

extern "C" void kernel_launch(void* const* d_in, const int* in_sizes, int n_in,
                              void* d_out, int out_size, void* d_ws, size_t ws_size,
                              hipStream_t stream) {
  (void)in_sizes; (void)n_in; (void)out_size; (void)ws_size;
  const float* x   = (const float*)d_in[0];
  const float* W1r = (const float*)d_in[1];
  const float* W1n = (const float*)d_in[2];
  const float* b1  = (const float*)d_in[3];
  const float* W2r = (const float*)d_in[4];
  const float* W2n = (const float*)d_in[5];
  const float* b2  = (const float*)d_in[6];
  const int* es0 = (const int*)d_in[7];
  const int* et0 = (const int*)d_in[8];
  const int* es1 = (const int*)d_in[9];
  const int* et1 = (const int*)d_in[10];

  float* ws   = (float*)d_ws;
  float* agg1 = ws + OFF_AGG1;
  float* cnt1 = ws + OFF_CNT1;
  float* agg2 = ws + OFF_AGG2;
  float* cnt2 = ws + OFF_CNT2;
  float* h    = ws + OFF_H;
  float* out  = (float*)d_out;

  // zero segment-sum accumulators + counters (graph-capture safe)
  hipMemsetAsync(agg1, 0, ((size_t)RES0 * N_FTRS + RES0) * sizeof(float), stream);
  hipMemsetAsync(agg2, 0, ((size_t)RES1 * N_HID + RES1) * sizeof(float), stream);

  sage_scatter1<<<NE0 / 8, 256, 0, stream>>>(x, es0, et0, agg1, cnt1);
  sage_invcnt<<<(RES0 + 255) / 256, 256, 0, stream>>>(cnt1, RES0);
  sage_gemm1<<<dim3(RES0 / 16, (N_HID + 15) / 16), 32, 0, stream>>>(
      x, agg1, cnt1, W1r, W1n, b1, h);
  sage_scatter2<<<NE1, 256, 0, stream>>>(h, es1, et1, agg2, cnt2);
  sage_invcnt<<<(RES1 + 255) / 256, 256, 0, stream>>>(cnt2, RES1);
  sage_gemm2<<<dim3(RES1 / 16, N_CLS / 16), 32, 0, stream>>>(
      h, agg2, cnt2, W2r, W2n, b2, out);
  sage_logsoftmax<<<RES1, 32, 0, stream>>>(out);
}